// GCNLPA_8830452760937
// MI455X (gfx1250) — compile-verified
//
#include <hip/hip_runtime.h>
#include <hip/hip_bf16.h>
#include <math.h>

// Problem constants (match reference)
#define N_NODES 10000
#define N_EDGES 320000
#define IN_DIM  128
#define HIDDEN  256
#define OUTC    40
#define LP_STEPS 3

typedef __attribute__((ext_vector_type(2))) float v2f;
typedef __attribute__((ext_vector_type(8))) float v8f;

// ---------------------------------------------------------------------------
// Elementwise / scatter helpers
// ---------------------------------------------------------------------------
__global__ void k_sigmoid(const float* __restrict__ ew, float* __restrict__ w, int n) {
    int i = blockIdx.x * blockDim.x + threadIdx.x;
    if (i < n) w[i] = 1.0f / (1.0f + __expf(-ew[i]));
}

__global__ void k_zero(float* __restrict__ p, int n) {
    int i = blockIdx.x * blockDim.x + threadIdx.x;
    if (i < n) p[i] = 0.0f;
}

__global__ void k_deg_edges(const int* __restrict__ dst, const float* __restrict__ w,
                            float* __restrict__ deg, int e) {
    int i = blockIdx.x * blockDim.x + threadIdx.x;
    if (i < e) atomicAdd(&deg[dst[i]], w[i]);
}

__global__ void k_dinv(const float* __restrict__ deg, float* __restrict__ dinv, int n) {
    int i = blockIdx.x * blockDim.x + threadIdx.x;
    if (i < n) dinv[i] = rsqrtf(deg[i] + 1.0f);   // +1 self-loop
}

// ---------------------------------------------------------------------------
// f32 WMMA GEMM:  C[M x NC] = A[M x K] @ B[K x NC]
// One wave32 per 16x16 C tile; A tile staged in padded LDS.
// M must be a multiple of 16 (10000 = 625*16).
// ---------------------------------------------------------------------------
template <int K, int NC>
__global__ __launch_bounds__(128) void k_gemm_wmma(const float* __restrict__ A,
                                                   const float* __restrict__ B,
                                                   float* __restrict__ C) {
    constexpr int KP = K + 1;                 // bank-conflict padding (K % 64 == 0)
    constexpr int TILES_N = (NC + 15) / 16;
    __shared__ float As[16 * KP];

    const int lane = threadIdx.x;             // 0..31 (wave32)
    const int wave = threadIdx.y;             // 0..3
    const int tm   = blockIdx.x;              // 16-row tile of A/C

    // cooperative, coalesced stage of the 16 x K A tile
    const float* Ablk = A + (size_t)tm * 16 * K;
    const int t = wave * 32 + lane;
    #pragma unroll
    for (int i = t; i < 16 * K; i += 128) {
        const int r = i / K, c = i - r * K;
        As[r * KP + c] = Ablk[i];
    }
    __syncthreads();

    const int tn = blockIdx.y * 4 + wave;
    if (tn >= TILES_N) return;                // whole-wave exit: EXEC stays all-1s for WMMA

    const int half = lane >> 4;               // 0: lanes 0-15, 1: lanes 16-31
    const int l16  = lane & 15;
    const int col  = tn * 16 + l16;
    const bool colOK = (NC % 16 == 0) || (col < NC);

    const float* Ar = As + l16 * KP;          // row M = l16 of the tile
    v8f acc = {};
    #pragma unroll 4
    for (int k = 0; k < K; k += 4) {
        v2f a, b;
        // A 16x4 f32 fragment: VGPR0/1 = K {0,1} (lanes 0-15) / {2,3} (lanes 16-31)
        a.x = Ar[k + 2 * half];
        a.y = Ar[k + 2 * half + 1];
        // B 4x16 fragment mirrors A: lane half selects K half, VGPR steps K
        if (colOK) {
            b.x = B[(size_t)(k + 2 * half)     * NC + col];
            b.y = B[(size_t)(k + 2 * half + 1) * NC + col];
        } else {
            b.x = 0.0f; b.y = 0.0f;
        }
        acc = __builtin_amdgcn_wmma_f32_16x16x4_f32(
            /*neg_a=*/false, a, /*neg_b=*/false, b,
            /*c_mod=*/(short)0, acc, /*reuse_a=*/false, /*reuse_b=*/false);
    }

    if (colOK) {
        // C/D layout: VGPR r -> row r + 8*half, col = l16
        const int row0 = tm * 16 + half * 8;
        #pragma unroll
        for (int r = 0; r < 8; ++r)
            C[(size_t)(row0 + r) * NC + col] = acc[r];
    }
}

// ---------------------------------------------------------------------------
// GCN aggregation: AGG = dinv^2 * H (self loop) + bias, then edge scatter
// ---------------------------------------------------------------------------
template <int F>
__global__ void k_selfinit(const float* __restrict__ H, const float* __restrict__ dinv,
                           const float* __restrict__ bias, float* __restrict__ AGG) {
    int i = blockIdx.x * blockDim.x + threadIdx.x;
    if (i >= N_NODES * F) return;
    const int row = i / F, c = i - row * F;
    const float d = dinv[row];
    AGG[i] = d * d * H[i] + bias[c];
}

template <int F>
__global__ void k_edge_agg(const float* __restrict__ H, float* __restrict__ AGG,
                           const int* __restrict__ src, const int* __restrict__ dst,
                           const float* __restrict__ w, const float* __restrict__ dinv) {
    constexpr int FB = F / 4;                 // float4 chunks per row
    int tid = blockIdx.x * blockDim.x + threadIdx.x;
    const int e = tid / FB, fb = tid - e * FB;
    if (e >= N_EDGES) return;
    const int s = src[e], d = dst[e];
    const float coef = dinv[s] * w[e] * dinv[d];
    const float4 v = *(const float4*)(H + (size_t)s * F + 4 * fb);
    float* o = AGG + (size_t)d * F + 4 * fb;
    atomicAdd(o + 0, coef * v.x);
    atomicAdd(o + 1, coef * v.y);
    atomicAdd(o + 2, coef * v.z);
    atomicAdd(o + 3, coef * v.w);
}

// ---------------------------------------------------------------------------
// BatchNorm (training stats, biased var) + ReLU
// ---------------------------------------------------------------------------
__global__ void k_bn_partial(const float* __restrict__ X, float* __restrict__ gsum,
                             float* __restrict__ gsq, int rowsPerBlock) {
    const int c  = threadIdx.x;               // one column per thread (F == 256)
    const int r0 = blockIdx.x * rowsPerBlock;
    const int r1 = min(r0 + rowsPerBlock, N_NODES);
    float s = 0.0f, q = 0.0f;
    for (int r = r0; r < r1; ++r) {
        const float v = X[(size_t)r * HIDDEN + c];
        s += v; q += v * v;
    }
    atomicAdd(&gsum[c], s);
    atomicAdd(&gsq[c], q);
}

__global__ void k_bn_final(const float* __restrict__ gsum, const float* __restrict__ gsq,
                           const float* __restrict__ g, const float* __restrict__ bt,
                           float* __restrict__ scale, float* __restrict__ shift) {
    const int c = threadIdx.x;
    const float m   = gsum[c] * (1.0f / N_NODES);
    const float var = gsq[c] * (1.0f / N_NODES) - m * m;
    const float sc  = g[c] * rsqrtf(var + 1e-5f);
    scale[c] = sc;
    shift[c] = bt[c] - m * sc;
}

__global__ void k_bn_relu(float* __restrict__ X, const float* __restrict__ scale,
                          const float* __restrict__ shift) {
    int i = blockIdx.x * blockDim.x + threadIdx.x;
    if (i >= N_NODES * HIDDEN) return;
    const int c = i & (HIDDEN - 1);
    X[i] = fmaxf(0.0f, X[i] * scale[c] + shift[c]);
}

// ---------------------------------------------------------------------------
// Row softmax (40 classes)
// ---------------------------------------------------------------------------
__global__ void k_softmax(const float* __restrict__ X, float* __restrict__ out) {
    int i = blockIdx.x * blockDim.x + threadIdx.x;
    if (i >= N_NODES) return;
    const float* r = X + (size_t)i * OUTC;
    float m = r[0];
    #pragma unroll
    for (int c = 1; c < OUTC; ++c) m = fmaxf(m, r[c]);
    float s = 0.0f;
    float e[OUTC];
    #pragma unroll
    for (int c = 0; c < OUTC; ++c) { e[c] = __expf(r[c] - m); s += e[c]; }
    const float inv = 1.0f / s;
    float* o = out + (size_t)i * OUTC;
    #pragma unroll
    for (int c = 0; c < OUTC; ++c) o[c] = e[c] * inv;
}

// ---------------------------------------------------------------------------
// Label propagation (sparse: A = scatter(w) + I, never materialized dense)
// ---------------------------------------------------------------------------
__global__ void k_onehot(const int* __restrict__ y, float* __restrict__ L) {
    int i = blockIdx.x * blockDim.x + threadIdx.x;
    if (i >= N_NODES * OUTC) return;
    const int row = i / OUTC, c = i - row * OUTC;
    L[i] = (y[row] == c) ? 1.0f : 0.0f;
}

__global__ void k_copy(const float* __restrict__ a, float* __restrict__ b, int n) {
    int i = blockIdx.x * blockDim.x + threadIdx.x;
    if (i < n) b[i] = a[i];
}

__global__ void k_lp_edge(const float* __restrict__ Lc, float* __restrict__ Ln,
                          const int* __restrict__ src, const int* __restrict__ dst,
                          const float* __restrict__ w) {
    int e = blockIdx.x * blockDim.x + threadIdx.x;
    if (e >= N_EDGES) return;
    const int s = src[e], d = dst[e];
    const float we = w[e];
    const float4* lr = (const float4*)(Lc + (size_t)d * OUTC);
    float* o = Ln + (size_t)s * OUTC;
    #pragma unroll
    for (int j = 0; j < OUTC / 4; ++j) {
        const float4 v = lr[j];
        atomicAdd(o + 4 * j + 0, we * v.x);
        atomicAdd(o + 4 * j + 1, we * v.y);
        atomicAdd(o + 4 * j + 2, we * v.z);
        atomicAdd(o + 4 * j + 3, we * v.w);
    }
}

__global__ void k_lp_norm(const float* __restrict__ L, float* __restrict__ out) {
    int i = blockIdx.x * blockDim.x + threadIdx.x;
    if (i >= N_NODES) return;
    const float* r = L + (size_t)i * OUTC;
    float s = 0.0f;
    #pragma unroll
    for (int c = 0; c < OUTC; ++c) s += r[c] * r[c];
    const float inv = 1.0f / fmaxf(sqrtf(s), 1e-12f);
    float* o = out + (size_t)i * OUTC;
    #pragma unroll
    for (int c = 0; c < OUTC; ++c) o[c] = r[c] * inv;
}

// ---------------------------------------------------------------------------
// Host-side orchestration (all launches on `stream`, graph-capture safe)
// ---------------------------------------------------------------------------
static inline int cdiv(long long a, int b) { return (int)((a + b - 1) / b); }

extern "C" void kernel_launch(void* const* d_in, const int* in_sizes, int n_in,
                              void* d_out, int out_size, void* d_ws, size_t ws_size,
                              hipStream_t stream) {
    (void)in_sizes; (void)n_in; (void)out_size; (void)ws_size;
    const float* x   = (const float*)d_in[0];
    const int*   ei  = (const int*)d_in[1];
    const int*   src = ei;
    const int*   dst = ei + N_EDGES;
    const int*   y   = (const int*)d_in[2];
    const float* ew  = (const float*)d_in[3];
    const float* W0  = (const float*)d_in[4];
    const float* b0  = (const float*)d_in[5];
    const float* W1  = (const float*)d_in[6];
    const float* b1  = (const float*)d_in[7];
    const float* W2  = (const float*)d_in[8];
    const float* b2  = (const float*)d_in[9];
    const float* g0  = (const float*)d_in[10];
    const float* bt0 = (const float*)d_in[11];
    const float* g1  = (const float*)d_in[12];
    const float* bt1 = (const float*)d_in[13];
    float* out1 = (float*)d_out;                       // softmax, N x 40
    float* out2 = out1 + (size_t)N_NODES * OUTC;       // normalized labels, N x 40

    // workspace layout (floats, 64B-aligned offsets)
    float* ws    = (float*)d_ws;
    float* wsig  = ws;                                 // E
    float* deg   = ws + 320000;                        // N
    float* dinv  = ws + 330000;                        // N
    float* scale = ws + 340000;                        // 256
    float* shift = ws + 340256;                        // 256
    float* gsum  = ws + 340512;                        // 256
    float* gsq   = ws + 340768;                        // 256
    float* H     = ws + 341024;                        // N*256
    float* AGG   = ws + 2901024;                       // N*256
    float* L0    = ws + 5461024;                       // N*40
    float* L1    = ws + 5861024;                       // N*40

    const dim3 gblk(32, 4);

    // edge weights + symmetric-norm degrees
    k_sigmoid<<<cdiv(N_EDGES, 256), 256, 0, stream>>>(ew, wsig, N_EDGES);
    k_zero<<<cdiv(N_NODES, 256), 256, 0, stream>>>(deg, N_NODES);
    k_deg_edges<<<cdiv(N_EDGES, 256), 256, 0, stream>>>(dst, wsig, deg, N_EDGES);
    k_dinv<<<cdiv(N_NODES, 256), 256, 0, stream>>>(deg, dinv, N_NODES);

    // ---- layer 1: x @ W0 -> aggregate -> BN -> ReLU ----
    k_gemm_wmma<IN_DIM, HIDDEN><<<dim3(N_NODES / 16, 4), gblk, 0, stream>>>(x, W0, H);
    k_selfinit<HIDDEN><<<cdiv((long long)N_NODES * HIDDEN, 256), 256, 0, stream>>>(H, dinv, b0, AGG);
    k_edge_agg<HIDDEN><<<cdiv((long long)N_EDGES * (HIDDEN / 4), 256), 256, 0, stream>>>(H, AGG, src, dst, wsig, dinv);
    k_zero<<<2, 256, 0, stream>>>(gsum, 512);          // gsum+gsq contiguous
    k_bn_partial<<<40, 256, 0, stream>>>(AGG, gsum, gsq, 250);
    k_bn_final<<<1, 256, 0, stream>>>(gsum, gsq, g0, bt0, scale, shift);
    k_bn_relu<<<cdiv((long long)N_NODES * HIDDEN, 256), 256, 0, stream>>>(AGG, scale, shift);

    // ---- layer 2 ----
    k_gemm_wmma<HIDDEN, HIDDEN><<<dim3(N_NODES / 16, 4), gblk, 0, stream>>>(AGG, W1, H);
    k_selfinit<HIDDEN><<<cdiv((long long)N_NODES * HIDDEN, 256), 256, 0, stream>>>(H, dinv, b1, AGG);
    k_edge_agg<HIDDEN><<<cdiv((long long)N_EDGES * (HIDDEN / 4), 256), 256, 0, stream>>>(H, AGG, src, dst, wsig, dinv);
    k_zero<<<2, 256, 0, stream>>>(gsum, 512);
    k_bn_partial<<<40, 256, 0, stream>>>(AGG, gsum, gsq, 250);
    k_bn_final<<<1, 256, 0, stream>>>(gsum, gsq, g1, bt1, scale, shift);
    k_bn_relu<<<cdiv((long long)N_NODES * HIDDEN, 256), 256, 0, stream>>>(AGG, scale, shift);

    // ---- layer 3: -> 40 classes -> softmax ----
    k_gemm_wmma<HIDDEN, OUTC><<<dim3(N_NODES / 16, 1), gblk, 0, stream>>>(AGG, W2, H);
    k_selfinit<OUTC><<<cdiv((long long)N_NODES * OUTC, 256), 256, 0, stream>>>(H, dinv, b2, AGG);
    k_edge_agg<OUTC><<<cdiv((long long)N_EDGES * (OUTC / 4), 256), 256, 0, stream>>>(H, AGG, src, dst, wsig, dinv);
    k_softmax<<<cdiv(N_NODES, 256), 256, 0, stream>>>(AGG, out1);

    // ---- sparse label propagation: labels = (S + I)^3 onehot(y), row-normalized ----
    k_onehot<<<cdiv((long long)N_NODES * OUTC, 256), 256, 0, stream>>>(y, L0);
    float* Lc = L0;
    float* Ln = L1;
    for (int step = 0; step < LP_STEPS; ++step) {
        k_copy<<<cdiv((long long)N_NODES * OUTC, 256), 256, 0, stream>>>(Lc, Ln, N_NODES * OUTC); // I term
        k_lp_edge<<<cdiv(N_EDGES, 256), 256, 0, stream>>>(Lc, Ln, src, dst, wsig);
        float* tmp = Lc; Lc = Ln; Ln = tmp;
    }
    k_lp_norm<<<cdiv(N_NODES, 256), 256, 0, stream>>>(Lc, out2);
}